// Reservoir_80075370266951
// MI455X (gfx1250) — compile-verified
//
#include <hip/hip_runtime.h>

// out[b,v] = sum_d R[b,v,d] * W[v,d]
// B=16384, N_V=40, Dr=300, fp32. Pure bandwidth-bound (AI ~0.5 F/B,
// roofline floor ~34us at 23.3 TB/s). Strategy: async global->LDS staging
// (global_load_async_to_lds_b128, ASYNCcnt), then V_WMMA_F32_16X16X4_F32
// with all-ones B as an fp32 reduction engine: per wave, a 16-row b-tile
// x one v; A[m,k] = R*W products, D rows = the 16 dot products.

#define B_TOT   16384
#define N_V     40
#define DR      300
#define CH      100            // K-chunk (floats)
#define CH4     (CH/4)         // 25 float4 per row-chunk
#define NCHUNK  (DR/CH)        // 3
#define WPB     8              // waves per block
#define ROWS    16             // b-tile rows per wave

typedef __attribute__((ext_vector_type(2))) float v2f;
typedef __attribute__((ext_vector_type(8))) float v8f;
typedef __attribute__((ext_vector_type(4))) int   v4i;

#if __has_builtin(__builtin_amdgcn_wmma_f32_16x16x4_f32)
#define USE_WMMA 1
#else
#define USE_WMMA 0
#endif

#if __has_builtin(__builtin_amdgcn_global_load_async_to_lds_b128)
#define USE_ASYNC 1
// clang's builtin signature (from diagnostics): (AS1 int4*, AS3 int4*, imm, imm)
typedef __attribute__((address_space(1))) v4i GInt4;
typedef __attribute__((address_space(3))) v4i LInt4;
#else
#define USE_ASYNC 0
#endif

__global__ __launch_bounds__(WPB * 32) void
Reservoir_80075370266951_kernel(const float* __restrict__ R,
                                const float* __restrict__ W,
                                float* __restrict__ out)
{
    // per-wave region: 16*CH R-tile + CH W-chunk = 1700 floats (6800 B, 16B aligned)
    __shared__ __align__(16) float lds[WPB * (ROWS * CH + CH)];

    const int tid  = threadIdx.x;
    const int wave = tid >> 5;
    const int lane = tid & 31;
    const int half = lane >> 4;     // which K-pair group this lane feeds
    const int row  = lane & 15;     // A-matrix row (b within tile)

    const int b0 = blockIdx.x * ROWS;
    const int v  = blockIdx.y * WPB + wave;

    float* Rl = lds + wave * (ROWS * CH + CH);
    float* Wl = Rl + ROWS * CH;

    const float4* Rg = (const float4*)R;   // row (b,v) starts at ((b*N_V+v)*DR)/4
    const float4* Wg = (const float4*)W;

#if USE_WMMA
    v8f acc = {};
    const v2f ones = {1.0f, 1.0f};
#else
    float acc_s = 0.0f;
#endif

    for (int chunk = 0; chunk < NCHUNK; ++chunk) {
        // ---- stage 16 rows x CH floats of R, 128b coalesced ----
        const long rowBase = ((long)b0 * N_V + v) * (DR / 4) + (long)chunk * CH4;
        for (int i = lane; i < ROWS * CH4; i += 32) {
            const int  r  = i / CH4;
            const int  cc = i - r * CH4;
            const long g  = rowBase + (long)r * (N_V * (DR / 4)) + cc;
#if USE_ASYNC
            __builtin_amdgcn_global_load_async_to_lds_b128(
                (GInt4*)(float4*)(Rg + g), (LInt4*)(Rl + i * 4), 0, 0);
#else
            *(float4*)(Rl + i * 4) = Rg[g];
#endif
        }
        // ---- stage W[v, chunk] ----
        if (lane < CH4) {
            const long gw = (long)v * (DR / 4) + (long)chunk * CH4 + lane;
#if USE_ASYNC
            __builtin_amdgcn_global_load_async_to_lds_b128(
                (GInt4*)(float4*)(Wg + gw), (LInt4*)(Wl + lane * 4), 0, 0);
#else
            *(float4*)(Wl + lane * 4) = Wg[gw];
#endif
        }

#if USE_ASYNC
#if __has_builtin(__builtin_amdgcn_s_wait_asynccnt)
        __builtin_amdgcn_s_wait_asynccnt(0);
#else
        asm volatile("s_wait_asynccnt 0" ::: "memory");
#endif
#endif
        __syncthreads();

        // ---- reduce CH columns via 25 x V_WMMA_F32_16X16X4_F32 ----
        // Lane feeds A[row, 2*half + {0,1}] = R*W; B = all-ones -> D rows = dot sums.
        const float* rbase = Rl + row * CH + 2 * half;
        const float* wbase = Wl + 2 * half;
#pragma unroll
        for (int k0 = 0; k0 < CH; k0 += 4) {
            v2f a = *(const v2f*)(rbase + k0);
            v2f w = *(const v2f*)(wbase + k0);
#if USE_WMMA
            a = a * w;
            acc = __builtin_amdgcn_wmma_f32_16x16x4_f32(
                false, a, false, ones, (short)0, acc, false, false);
#else
            acc_s += a.x * w.x + a.y * w.y;
#endif
        }
        __syncthreads();
    }

#if USE_WMMA
    // D layout: lane 0 holds M=0..7 in acc[0..7]; lane 16 holds M=8..15.
    // All 16 columns are identical (B was all-ones).
    if (lane == 0 || lane == 16) {
        const int mb = (lane >> 4) * 8;
#pragma unroll
        for (int j = 0; j < 8; ++j)
            out[(long)(b0 + mb + j) * N_V + v] = acc[j];
    }
#else
    acc_s += __shfl_down(acc_s, 16, 32);
    if (lane < 16)
        out[(long)(b0 + lane) * N_V + v] = acc_s;
#endif
}

extern "C" void kernel_launch(void* const* d_in, const int* in_sizes, int n_in,
                              void* d_out, int out_size, void* d_ws, size_t ws_size,
                              hipStream_t stream) {
    const float* R = (const float*)d_in[0];   // [16384, 40, 300] fp32
    const float* W = (const float*)d_in[1];   // [40, 300] fp32
    float* out = (float*)d_out;               // [16384, 40] fp32

    dim3 grid(B_TOT / ROWS, N_V / WPB);       // (1024, 5)
    dim3 block(WPB * 32);                     // 256 threads = 8 waves
    Reservoir_80075370266951_kernel<<<grid, block, 0, stream>>>(R, W, out);
}